// UKFCell_76355928588354
// MI455X (gfx1250) — compile-verified
//
#include <hip/hip_runtime.h>

// CDNA5 / gfx1250: wave32, WMMA 16x16x4 F32.
// One wave processes one batch element (16x16 UKF collapsed to Kalman update).

typedef float v2f __attribute__((ext_vector_type(2)));
typedef float v8f __attribute__((ext_vector_type(8)));

__device__ __forceinline__ float rlane(float v, int lane) {
    return __uint_as_float(__builtin_amdgcn_readlane(__float_as_uint(v), lane));
}

__global__ __launch_bounds__(256)
void ukf_kernel(const float* __restrict__ meas,
                const float* __restrict__ state,
                const float* __restrict__ cov,
                const float* __restrict__ pnoise,
                const float* __restrict__ mnoise,
                float* __restrict__ out_y,
                float* __restrict__ out_x,
                float* __restrict__ out_P,
                int Btot) {
    __shared__ __align__(16) float lds[8 * 512];   // per wave: 256 (A) + 256 (S)

    const int tid = threadIdx.x;
    const int wid = tid >> 5;          // wave id in block
    const int l   = tid & 31;          // lane
    const int hi  = l >> 4;            // half-wave (K / M split)
    const int row = l & 15;            // matrix row owned by this lane
    const int batch = blockIdx.x * 8 + wid;
    if (batch >= Btot) return;         // wave-uniform, EXEC stays full inside

    // ---------------- packed lower-tri noise factors -> WMMA A-layout -------
    // lane l holds row `row`; VGPR pair (q, r) holds column 4q + r + 2*hi.
    const float* pnb = pnoise + (size_t)batch * 136;
    const float* mnb = mnoise + (size_t)batch * 136;
    const int tribase = (row * (row + 1)) >> 1;
    float lp[8], lm[8];
#pragma unroll
    for (int q = 0; q < 4; ++q) {
#pragma unroll
        for (int r = 0; r < 2; ++r) {
            const int col = 4 * q + r + 2 * hi;
            float vp = 0.0f, vm = 0.0f;
            if (col <= row) {                       // lower triangle only
                vp = pnb[tribase + col];
                vm = mnb[tribase + col];
            }
            lp[2 * q + r] = vp;
            lm[2 * q + r] = vm;
        }
    }

    // ---------------- PQ = Lp*Lp^T, R = Lm*Lm^T  (8x v_wmma_f32_16x16x4_f32)
    // Symmetric product: A-operand registers double as B-operand registers.
    v8f pq = {};
    v8f rr = {};
#pragma unroll
    for (int q = 0; q < 4; ++q) {
        v2f a; a.x = lp[2 * q]; a.y = lp[2 * q + 1];
        pq = __builtin_amdgcn_wmma_f32_16x16x4_f32(false, a, false, a,
                                                   (short)0, pq, false, false);
        v2f b; b.x = lm[2 * q]; b.y = lm[2 * q + 1];
        rr = __builtin_amdgcn_wmma_f32_16x16x4_f32(false, b, false, b,
                                                   (short)0, rr, false, false);
    }

    // ---------------- A = P + PQ, S = A + R  in WMMA C-layout ----------------
    const float* Pb = cov + (size_t)batch * 256;
    v8f Ac, Sc;
#pragma unroll
    for (int r = 0; r < 8; ++r) {
        const int m = r + 8 * hi;
        const float a = Pb[m * 16 + row] + pq[r];
        Ac[r] = a;
        Sc[r] = a + rr[r];
    }

    // ---------------- C-layout -> row-in-lane via per-wave LDS slice ---------
    float* wbuf = lds + wid * 512;
#pragma unroll
    for (int r = 0; r < 8; ++r) {
        const int m = r + 8 * hi;
        wbuf[m * 16 + row]       = Ac[r];
        wbuf[256 + m * 16 + row] = Sc[r];
    }
    asm volatile("s_wait_dscnt 0" ::: "memory");   // same-wave LDS RAW fence

    float arow[16], s[16];
#pragma unroll
    for (int j4 = 0; j4 < 4; ++j4) {
        const float4 a4 = *(const float4*)(wbuf + row * 16 + 4 * j4);
        const float4 s4 = *(const float4*)(wbuf + 256 + row * 16 + 4 * j4);
        arow[4 * j4 + 0] = a4.x; arow[4 * j4 + 1] = a4.y;
        arow[4 * j4 + 2] = a4.z; arow[4 * j4 + 3] = a4.w;
        s[4 * j4 + 0] = s4.x; s[4 * j4 + 1] = s4.y;
        s[4 * j4 + 2] = s4.z; s[4 * j4 + 3] = s4.w;
    }

    // ---------------- Cholesky S = L*L^T (row-in-lane, readlane broadcasts) --
    float dinv[16];                                 // 1/L_kk, wave-uniform
#pragma unroll
    for (int k = 0; k < 16; ++k) {
        const float p  = rlane(s[k], k);
        float inv = __frsqrt_rn(p);
        inv = inv * (1.5f - 0.5f * p * inv * inv);  // Newton refine
        s[k] *= inv;                                // column k of L
        dinv[k] = inv;                              // 1/sqrt(p) == 1/L_kk
#pragma unroll
        for (int j = k + 1; j < 16; ++j) {
            const float ljk = rlane(s[k], j);
            s[j] = fmaf(-ljk, s[k], s[j]);          // rank-1 trailing update
        }
    }

    // ---------------- solve S * X = A  ->  lane c holds gain row c -----------
    // forward: L y = b   (b = column c of A == row c, A symmetric)
    float yv[16];
#pragma unroll
    for (int i = 0; i < 16; ++i) {
        float t = arow[i];
#pragma unroll
        for (int j = 0; j < i; ++j)
            t = fmaf(-rlane(s[j], i), yv[j], t);    // L[i][j]
        yv[i] = t * dinv[i];
    }
    // backward: L^T x = y
    float xv[16];
#pragma unroll
    for (int i = 15; i >= 0; --i) {
        float t = yv[i];
#pragma unroll
        for (int j = i + 1; j < 16; ++j)
            t = fmaf(-rlane(s[i], j), xv[j], t);    // L[j][i]
        xv[i] = t * dinv[i];
    }

    // ---------------- state update: new_x = mu + gain*(z - mu), y = mu -------
    const float vmu = state[(size_t)batch * 16 + row];
    const float vz  = meas [(size_t)batch * 16 + row];
    const float d   = vz - vmu;                     // lane j holds d_j
    float upd = 0.0f;
#pragma unroll
    for (int j = 0; j < 16; ++j)
        upd = fmaf(xv[j], rlane(d, j), upd);        // gain[row]·d
    if (hi == 0) {
        out_y[(size_t)batch * 16 + row] = vmu;      // y == x == mu (exact)
        out_x[(size_t)batch * 16 + row] = vmu + upd;
    }

    // ---------------- new_cov = A - gain*A  (4x WMMA, C preloaded with A) ----
    // A-operand: rows of (-gain); B-operand: A (symmetric -> reuse arow).
    v8f nc = Ac;
#pragma unroll
    for (int q = 0; q < 4; ++q) {
        v2f ga, ab;
        ga.x = -(hi ? xv[4 * q + 2] : xv[4 * q + 0]);
        ga.y = -(hi ? xv[4 * q + 3] : xv[4 * q + 1]);
        ab.x =  (hi ? arow[4 * q + 2] : arow[4 * q + 0]);
        ab.y =  (hi ? arow[4 * q + 3] : arow[4 * q + 1]);
        nc = __builtin_amdgcn_wmma_f32_16x16x4_f32(false, ga, false, ab,
                                                   (short)0, nc, false, false);
    }
    float* Pout = out_P + (size_t)batch * 256;
#pragma unroll
    for (int r = 0; r < 8; ++r) {
        const int m = r + 8 * hi;
        Pout[m * 16 + row] = nc[r];
    }
}

extern "C" void kernel_launch(void* const* d_in, const int* in_sizes, int n_in,
                              void* d_out, int out_size, void* d_ws, size_t ws_size,
                              hipStream_t stream) {
    (void)n_in; (void)out_size; (void)d_ws; (void)ws_size;
    const float* meas   = (const float*)d_in[0];   // (B,16)
    const float* state  = (const float*)d_in[1];   // (B,16)
    const float* cov    = (const float*)d_in[2];   // (B,16,16)
    const float* pnoise = (const float*)d_in[3];   // (B,136)
    const float* mnoise = (const float*)d_in[4];   // (B,136)
    const int Btot = in_sizes[1] / 16;

    float* out   = (float*)d_out;
    float* out_y = out;                            // (B,16)
    float* out_x = out + (size_t)Btot * 16;        // (B,16)
    float* out_P = out + (size_t)Btot * 32;        // (B,16,16)

    const int blocks = (Btot + 7) / 8;             // 8 waves (batch elems) / block
    ukf_kernel<<<blocks, 256, 0, stream>>>(meas, state, cov, pnoise, mnoise,
                                           out_y, out_x, out_P, Btot);
}